// GCN_79843442032814
// MI455X (gfx1250) — compile-verified
//
#include <hip/hip_runtime.h>

typedef __attribute__((ext_vector_type(2))) float v2f;
typedef __attribute__((ext_vector_type(8))) float v8f;

// ---------------------------------------------------------------------------
// degree / normalization
// ---------------------------------------------------------------------------
__global__ void k_deg_init(float* __restrict__ deg, int n) {
  int i = blockIdx.x * blockDim.x + threadIdx.x;
  if (i < n) deg[i] = 1.0f;  // self-loop contributes 1 to every node
}

__global__ void k_deg_acc(const long long* __restrict__ col,
                          float* __restrict__ deg, int E) {
  int e = blockIdx.x * blockDim.x + threadIdx.x;
  if (e < E) atomicAdd(&deg[(int)col[e]], 1.0f);
}

__global__ void k_rsqrt_inplace(float* __restrict__ deg, int n) {
  int i = blockIdx.x * blockDim.x + threadIdx.x;
  if (i < n) deg[i] = rsqrtf(deg[i]);  // deg >= 1 always (self-loops)
}

// ---------------------------------------------------------------------------
// Layer 1 dense transform: h1[n,16] = x[n,9] @ W1[9,16] via V_WMMA_F32_16X16X4_F32
// One wave32 handles a 16-node tile. K padded 9 -> 12 (3 k-steps of 4).
// A 16x4 f32 layout: lanes 0-15 hold K={0,1}, lanes 16-31 hold K={2,3}.
// B 4x16 f32 layout: VGPR0 = rows K0/K2 across lane halves, VGPR1 = K1/K3.
// D 16x16 f32 layout: VGPR v -> M=v (lanes<16) / M=v+8 (lanes>=16), N=lane&15.
// ---------------------------------------------------------------------------
__global__ void k_lin1_wmma(const float* __restrict__ x,
                            const float* __restrict__ W1,
                            float* __restrict__ h1, int n) {
  const int lane = threadIdx.x & 31;
  const int wave = threadIdx.x >> 5;
  const int gw   = blockIdx.x * (blockDim.x >> 5) + wave;
  const int node_base = gw * 16;
  if (node_base >= n) return;            // wave-uniform exit: EXEC all-1 at WMMA

  const int m    = lane & 15;
  const int koff = (lane < 16) ? 0 : 2;
  const long long node = node_base + m;

  v8f c = {};
#pragma unroll
  for (int ks = 0; ks < 3; ++ks) {
    const int k0 = ks * 4 + koff;
    v2f a, b;
    a.x = (k0     < 9) ? x[node * 9 + k0]       : 0.0f;
    a.y = (k0 + 1 < 9) ? x[node * 9 + k0 + 1]   : 0.0f;
    b.x = (k0     < 9) ? W1[k0 * 16 + m]        : 0.0f;
    b.y = (k0 + 1 < 9) ? W1[(k0 + 1) * 16 + m]  : 0.0f;
    c = __builtin_amdgcn_wmma_f32_16x16x4_f32(false, a, false, b,
                                              (short)0, c, false, false);
  }

  const int mrow = (lane < 16) ? 0 : 8;
#pragma unroll
  for (int v = 0; v < 8; ++v)
    h1[(long long)(node_base + mrow + v) * 16 + m] = c[v];
}

// ---------------------------------------------------------------------------
// Layer 2 dense transform with fused bias+ReLU on the input:
// h2[n,10] = relu(agg1[n,16] + b1) @ W2[16,10], K=16 (4 k-steps), N padded to 16.
// ---------------------------------------------------------------------------
__global__ void k_lin2_wmma(const float* __restrict__ agg1,
                            const float* __restrict__ b1,
                            const float* __restrict__ W2,
                            float* __restrict__ h2, int n) {
  const int lane = threadIdx.x & 31;
  const int wave = threadIdx.x >> 5;
  const int gw   = blockIdx.x * (blockDim.x >> 5) + wave;
  const int node_base = gw * 16;
  if (node_base >= n) return;

  const int m    = lane & 15;
  const int koff = (lane < 16) ? 0 : 2;
  const long long node = node_base + m;

  v8f c = {};
#pragma unroll
  for (int ks = 0; ks < 4; ++ks) {
    const int k0 = ks * 4 + koff;
    v2f a, b;
    float a0 = agg1[node * 16 + k0]     + b1[k0];
    float a1 = agg1[node * 16 + k0 + 1] + b1[k0 + 1];
    a.x = a0 > 0.0f ? a0 : 0.0f;
    a.y = a1 > 0.0f ? a1 : 0.0f;
    b.x = (m < 10) ? W2[k0 * 10 + m]       : 0.0f;
    b.y = (m < 10) ? W2[(k0 + 1) * 10 + m] : 0.0f;
    c = __builtin_amdgcn_wmma_f32_16x16x4_f32(false, a, false, b,
                                              (short)0, c, false, false);
  }

  const int mrow = (lane < 16) ? 0 : 8;
  if (m < 10) {
#pragma unroll
    for (int v = 0; v < 8; ++v)
      h2[(long long)(node_base + mrow + v) * 10 + m] = c[v];
  }
}

// ---------------------------------------------------------------------------
// Self-loop seed: agg[i][f] = h[i][f] * dis[i]^2  (also zero-initializes agg)
// ---------------------------------------------------------------------------
template <int F>
__global__ void k_self_seed(const float* __restrict__ h,
                            const float* __restrict__ dis,
                            float* __restrict__ agg, int n) {
  int t = blockIdx.x * blockDim.x + threadIdx.x;
  if (t < n * F) {
    int i = t / F;
    float d = dis[i];
    agg[t] = h[t] * d * d;
  }
}

// ---------------------------------------------------------------------------
// Edge aggregation: one thread per edge, F-float row gather + F atomic adds.
// ---------------------------------------------------------------------------
template <int F>
__global__ void k_edge_agg(const long long* __restrict__ row,
                           const long long* __restrict__ col,
                           const float* __restrict__ dis,
                           const float* __restrict__ h,
                           float* __restrict__ agg, int E) {
  int e = blockIdx.x * blockDim.x + threadIdx.x;
  if (e >= E) return;
  int r = (int)row[e];
  int c = (int)col[e];
  float nrm = dis[r] * dis[c];
  const float* hr = h + (long long)r * F;
  float* ac = agg + (long long)c * F;
#pragma unroll
  for (int f = 0; f < F; ++f)
    atomicAdd(&ac[f], hr[f] * nrm);
}

// ---------------------------------------------------------------------------
// Readout: out[i] = sum_f (agg2[i][f] + b2[f]) * Wf[f] + bf
// ---------------------------------------------------------------------------
__global__ void k_final(const float* __restrict__ agg2,
                        const float* __restrict__ b2,
                        const float* __restrict__ Wf,
                        const float* __restrict__ bf,
                        float* __restrict__ out, int n) {
  int i = blockIdx.x * blockDim.x + threadIdx.x;
  if (i >= n) return;
  float s = bf[0];
#pragma unroll
  for (int f = 0; f < 10; ++f)
    s += (agg2[(long long)i * 10 + f] + b2[f]) * Wf[f];
  out[i] = s;
}

// ---------------------------------------------------------------------------
// launch
// ---------------------------------------------------------------------------
static inline size_t align256(size_t x) { return (x + 255) & ~size_t(255); }

extern "C" void kernel_launch(void* const* d_in, const int* in_sizes, int n_in,
                              void* d_out, int out_size, void* d_ws, size_t ws_size,
                              hipStream_t stream) {
  const float*      x   = (const float*)d_in[0];
  const long long*  ei  = (const long long*)d_in[1];   // int64 [2, E]
  const float*      W1  = (const float*)d_in[2];
  const float*      b1  = (const float*)d_in[3];
  const float*      W2  = (const float*)d_in[4];
  const float*      b2  = (const float*)d_in[5];
  const float*      Wf  = (const float*)d_in[6];
  const float*      bf  = (const float*)d_in[7];
  float*            out = (float*)d_out;

  const int n = in_sizes[0] / 9;
  const int E = in_sizes[1] / 2;
  const long long* row = ei;
  const long long* col = ei + E;

  // Workspace carve (~132 MB): dis[n] | buf1[n*16] (h1 then h2) | buf2[n*16] (agg1 then agg2)
  char* ws = (char*)d_ws;
  float* dis  = (float*)ws;
  float* buf1 = (float*)(ws + align256((size_t)n * 4));
  float* buf2 = (float*)(ws + align256((size_t)n * 4) + align256((size_t)n * 16 * 4));

  const int B = 256;
  const int gN   = (n + B - 1) / B;
  const int gE   = (E + B - 1) / B;
  const int g16N = (n * 16 + B - 1) / B;   // n*16 = 16M fits in int
  const int g10N = (n * 10 + B - 1) / B;
  const int waves = (n + 15) / 16;         // one wave per 16-node tile
  const int gW    = (waves + 3) / 4;       // 128 threads = 4 waves per block

  // 1) degree + normalization
  k_deg_init<<<gN, B, 0, stream>>>(dis, n);
  k_deg_acc<<<gE, B, 0, stream>>>(col, dis, E);
  k_rsqrt_inplace<<<gN, B, 0, stream>>>(dis, n);

  // 2) layer 1: dense WMMA transform, self-loop seed, edge scatter
  k_lin1_wmma<<<gW, 128, 0, stream>>>(x, W1, buf1, n);
  k_self_seed<16><<<g16N, B, 0, stream>>>(buf1, dis, buf2, n);
  k_edge_agg<16><<<gE, B, 0, stream>>>(row, col, dis, buf1, buf2, E);

  // 3) layer 2: fused bias+relu into WMMA transform, seed, edge scatter
  k_lin2_wmma<<<gW, 128, 0, stream>>>(buf2, b1, W2, buf1, n);
  k_self_seed<10><<<g10N, B, 0, stream>>>(buf1, dis, buf2, n);
  k_edge_agg<10><<<gE, B, 0, stream>>>(row, col, dis, buf1, buf2, E);

  // 4) readout
  k_final<<<gN, B, 0, stream>>>(buf2, b2, Wf, bf, out, n);
}